// RibonanzaNetTriangleProjection_17325898072625
// MI455X (gfx1250) — compile-verified
//
#include <hip/hip_runtime.h>
#include <hip/hip_bf16.h>

// Problem constants (match reference)
#define HID 256
#define PA  32
#define PW  64
#define BB  4
#define LL  512

typedef float v2f __attribute__((ext_vector_type(2)));
typedef float v8f __attribute__((ext_vector_type(8)));

// ---------------- Stage 1: h[b,l,p] = hidden[b,l,:]·W_in[p,:] + b_in[p] ----
// [B*L, 32] rows; block (32,8): x = p, y = row-in-block. 33 MFLOP, trivial.
__global__ __launch_bounds__(256)
void tri_inproj_kernel(const float* __restrict__ hidden,
                       const float* __restrict__ W_in,
                       const float* __restrict__ b_in,
                       float* __restrict__ h) {
    const int p   = threadIdx.x;                    // 0..31
    const int row = blockIdx.x * 8 + threadIdx.y;   // 0..B*L-1
    const float* hr = hidden + (size_t)row * HID;
    const float* wr = W_in   + (size_t)p   * HID;
    float acc = b_in[p];
#pragma unroll 8
    for (int k = 0; k < HID; ++k) acc = __builtin_fmaf(hr[k], wr[k], acc);
    h[(size_t)row * PA + p] = acc;
}

// ---------------- Stage 2: t[b,j,o,q] = sum_p h[b,j,p] * W_out[o, p*PA+q] --
// One block per (b,j); 256 threads, each computes 8 of the 64*32 = 2048
// (o,q) elements. Consecutive threads -> consecutive q -> coalesced W_out.
__global__ __launch_bounds__(256)
void tri_mid_kernel(const float* __restrict__ h,
                    const float* __restrict__ W_out,
                    float* __restrict__ t) {
    __shared__ float hs[PA];
    const int bj  = blockIdx.x;      // b*L + j
    const int tid = threadIdx.x;
    if (tid < PA) hs[tid] = h[(size_t)bj * PA + tid];
    __syncthreads();
#pragma unroll
    for (int e = 0; e < 8; ++e) {
        const int idx = e * 256 + tid;          // 0..2047
        const int o = idx >> 5, q = idx & 31;
        const float* w = W_out + (size_t)o * (PA * PA) + q;
        float acc = 0.f;
#pragma unroll
        for (int p = 0; p < PA; ++p) acc = __builtin_fmaf(hs[p], w[p * PA], acc);
        t[(size_t)bj * (PW * PA) + idx] = acc;
    }
}

// ---------------- Stage 3 (memory-bound main loop, fp32 WMMA) --------------
// One wave per 16x16 output tile: rows = 16 i's, cols = 16 o's, fixed (b,j).
// C[i,o] = sum_q h[b,i,q] * t[b,j,o,q]   (K = 32 = 8 x v_wmma_f32_16x16x4_f32)
// then fuse + b_out[o] + pairwise[b,i,j,o] and stream the 268 MB out once.
__global__ __launch_bounds__(256, 2)
void tri_main_kernel(const float* __restrict__ h,
                     const float* __restrict__ t,
                     const float* __restrict__ b_out,
                     const float* __restrict__ pairwise,
                     float* __restrict__ out) {
    // wave id -> tile coordinates
    const unsigned w    = blockIdx.x * 8u + threadIdx.y;
    const unsigned ot   = w & 3u;            // o-tile   (PW/16 = 4)
    const unsigned j    = (w >> 2) & (LL - 1u);
    const unsigned it   = (w >> 11) & 31u;   // i-tile   (LL/16 = 32)
    const unsigned b    = w >> 16;           // batch
    const unsigned lane = threadIdx.x;       // wave32
    const unsigned l16  = lane & 15u;
    const unsigned half = lane >> 4;

    // A tile: h[b, i0+l16, q], per K-step lanes 0-15 hold K={0,1}, 16-31 K={2,3}
    // -> per-lane float2 at q = 4*kk + 2*half. B tile from t has the same
    //    per-lane pattern (q is the fastest-varying dim of t).
    const float* ha = h + ((size_t)(b * LL + it * 16 + l16)) * PA + 2 * half;
    const float* tb = t + (((size_t)(b * LL + j)) * PW + ot * 16 + l16) * PA + 2 * half;

    v2f A[8], Bm[8];
#pragma unroll
    for (int kk = 0; kk < 8; ++kk) {
        A[kk]  = *(const v2f*)(ha + 4 * kk);
        Bm[kk] = *(const v2f*)(tb + 4 * kk);
    }

    v8f c = {0.f, 0.f, 0.f, 0.f, 0.f, 0.f, 0.f, 0.f};
#pragma unroll
    for (int kk = 0; kk < 8; ++kk) {
        // (neg_a, A, neg_b, B, c_mod, C, reuse_a, reuse_b)
        c = __builtin_amdgcn_wmma_f32_16x16x4_f32(
                false, A[kk], false, Bm[kk], (short)0, c, false, false);
    }

    // Epilogue: C VGPR r -> i = i0 + r + 8*half ; lane l16 -> o = o0 + l16.
    const unsigned o  = ot * 16 + l16;
    const float bias  = b_out[o];
    const unsigned i0 = it * 16 + half * 8;
    const size_t base = ((((size_t)b * LL + i0) * LL + j) * PW) + o;
    const size_t strideI = (size_t)LL * PW;       // between consecutive i
    const float* pw = pairwise + base;
    float*       op = out      + base;
#pragma unroll
    for (int r = 0; r < 8; ++r) {
        op[(size_t)r * strideI] = c[r] + bias + pw[(size_t)r * strideI];
    }
}

extern "C" void kernel_launch(void* const* d_in, const int* in_sizes, int n_in,
                              void* d_out, int out_size, void* d_ws, size_t ws_size,
                              hipStream_t stream) {
    const float* hidden   = (const float*)d_in[0]; // [B,L,HID]
    const float* pairwise = (const float*)d_in[1]; // [B,L,L,PW]
    const float* W_in     = (const float*)d_in[2]; // [PA,HID]
    const float* b_in     = (const float*)d_in[3]; // [PA]
    const float* W_out    = (const float*)d_in[4]; // [PW,PA*PA]
    const float* b_out    = (const float*)d_in[5]; // [PW]
    float*       outp     = (float*)d_out;         // [B,L,L,PW]

    // Scratch layout: h [B*L*PA] floats, then t [B*L*PW*PA] floats (~16.25 MB)
    float* h_ws = (float*)d_ws;
    float* t_ws = h_ws + (size_t)BB * LL * PA;

    // Stage 1: in-projection
    tri_inproj_kernel<<<(BB * LL) / 8, dim3(32, 8), 0, stream>>>(hidden, W_in, b_in, h_ws);

    // Stage 2: t = einsum("bjp,opq->bjoq", h, W_out)
    tri_mid_kernel<<<BB * LL, 256, 0, stream>>>(h_ws, W_out, t_ws);

    // Stage 3: WMMA outer product + bias + residual, streamed once
    // waves = B * (L/16) * L * (PW/16) = 262144 ; 8 waves / block
    tri_main_kernel<<<(BB * (LL / 16) * LL * (PW / 16)) / 8, dim3(32, 8), 0, stream>>>(
        h_ws, t_ws, b_out, pairwise, outp);
}